// CustomLoss_71597104824804
// MI455X (gfx1250) — compile-verified
//
#include <hip/hip_runtime.h>
#include <hip/hip_bf16.h>

typedef float v2f __attribute__((ext_vector_type(2)));
typedef float v8f __attribute__((ext_vector_type(8)));

#define NCLASS 32
#define ROWS_PER_THREAD 8
#define BLOCK 256

// Exact f32 wave32 sum using V_WMMA_F32_16X16X4_F32.
// A: VGPR0 holds lanes 0-15 -> (M=lane,K=0), lanes 16-31 -> (M=lane-16,K=2);
// VGPR1 (K=1/K=3) = 0. B = all ones => D[m][n] = v_m + v_{m+16} (replicated over n).
// Each lane then holds rows {0..7} (lanes 0-15) or {8..15} (lanes 16-31) in its
// 8 D VGPRs; summing them + one xor-16 exchange yields the full 32-lane sum.
__device__ __forceinline__ float wave_sum_wmma(float x) {
    v2f a; a.x = x;    a.y = 0.0f;
    v2f b; b.x = 1.0f; b.y = 1.0f;
    v8f c = {};
    v8f d = __builtin_amdgcn_wmma_f32_16x16x4_f32(
        /*neg_a=*/false, a, /*neg_b=*/false, b,
        /*c_mod=*/(short)0, c, /*reuse_a=*/false, /*reuse_b=*/false);
    float s = ((d[0] + d[1]) + (d[2] + d[3])) + ((d[4] + d[5]) + (d[6] + d[7]));
    s += __shfl_xor(s, 16, 32);   // combine the two half-wave row groups
    return s;                     // every lane: sum of all 32 lanes
}

__global__ void __launch_bounds__(BLOCK)
ce_partial_kernel(const float* __restrict__ logits,
                  const int*   __restrict__ target,
                  float* __restrict__ partial,
                  int*   __restrict__ pflag,
                  long long nrows, long long stride) {
    __shared__ float wsum[BLOCK / 32];
    __shared__ int   bflag;
    if (threadIdx.x == 0) bflag = 0;
    __syncthreads();

    const float L2E = 1.4426950408889634f;   // log2(e)
    const float LN2 = 0.6931471805599453f;   // ln(2)

    long long tid = (long long)blockIdx.x * blockDim.x + threadIdx.x;
    float acc = 0.0f;
    int   fl  = 0;

    for (int r = 0; r < ROWS_PER_THREAD; ++r) {
        long long row = tid + (long long)r * stride;
        if (row < nrows) {
            const float4* rp = (const float4*)(logits + row * NCLASS);
            // speculative prefetch of next strided row (gfx1250 global_prefetch_b8)
            __builtin_prefetch(logits + (row + stride) * NCLASS, 0, 0);

            float v[NCLASS];
#pragma unroll
            for (int j = 0; j < NCLASS / 4; ++j) {
                float4 q = rp[j];
                v[4 * j + 0] = q.x; v[4 * j + 1] = q.y;
                v[4 * j + 2] = q.z; v[4 * j + 3] = q.w;
            }
            int t = target[row];

            float m = v[0];
#pragma unroll
            for (int i = 1; i < NCLASS; ++i) m = fmaxf(m, v[i]);

            float picked = 0.0f;
#pragma unroll
            for (int i = 0; i < NCLASS; ++i) picked = (i == t) ? v[i] : picked;

            float mb = m * L2E;
            float s = 0.0f;
#pragma unroll
            for (int i = 0; i < NCLASS; ++i)
                s += __builtin_amdgcn_exp2f(__builtin_fmaf(v[i], L2E, -mb));

            acc += m + LN2 * __builtin_amdgcn_logf(s) - picked;

            // argmax(row) == 3 (first-index semantics): v[3] equals the max and
            // strictly beats v[0..2].
            int am3 = (v[3] >= m) & (v[3] > v[0]) & (v[3] > v[1]) & (v[3] > v[2]);
            fl |= (t == 2) & am3;
        }
    }

    // exact f32 wave reduction via WMMA (uniform call, EXEC all ones)
    float w = wave_sum_wmma(acc);
    int lane = threadIdx.x & 31;
    int wid  = threadIdx.x >> 5;
    if (lane == 0) wsum[wid] = w;
    atomicOr(&bflag, fl);   // LDS ds_or; order-independent
    __syncthreads();

    if (threadIdx.x == 0) {
        float tot = 0.0f;
#pragma unroll
        for (int i = 0; i < BLOCK / 32; ++i) tot += wsum[i];
        partial[blockIdx.x] = tot;
        pflag[blockIdx.x]   = bflag;
    }
}

__global__ void __launch_bounds__(BLOCK)
ce_final_kernel(const float* __restrict__ partial,
                const int*   __restrict__ pflag,
                int nparts,
                const int* __restrict__ epoch_p,
                float* __restrict__ out,
                long long nrows) {
    __shared__ float wsum[BLOCK / 32];
    __shared__ int   bflag;
    if (threadIdx.x == 0) bflag = 0;
    __syncthreads();

    float acc = 0.0f;
    int   fl  = 0;
    for (int i = threadIdx.x; i < nparts; i += blockDim.x) {
        acc += partial[i];
        fl  |= pflag[i];
    }

    float w = wave_sum_wmma(acc);
    int lane = threadIdx.x & 31;
    int wid  = threadIdx.x >> 5;
    if (lane == 0) wsum[wid] = w;
    atomicOr(&bflag, fl);
    __syncthreads();

    if (threadIdx.x == 0) {
        float tot = 0.0f;
#pragma unroll
        for (int i = 0; i < BLOCK / 32; ++i) tot += wsum[i];
        float mean = tot / (float)nrows;                      // init_loss
        float e    = (float)epoch_p[0];
        float corr = powf(e, -0.65f) * 64.0f + 0.01f;         // correction_increase
        float loss = mean + (bflag ? corr : 0.0f);
        bool  bad  = (loss < 0.0f) || ((loss / mean) < 0.2f);
        out[0] = bad ? mean : loss;
    }
}

extern "C" void kernel_launch(void* const* d_in, const int* in_sizes, int n_in,
                              void* d_out, int out_size, void* d_ws, size_t ws_size,
                              hipStream_t stream) {
    const float* logits = (const float*)d_in[0];
    const int*   target = (const int*)d_in[1];
    const int*   epoch  = (const int*)d_in[2];

    long long nrows = (long long)in_sizes[0] / NCLASS;      // 2,097,152
    long long totalThreads = (nrows + ROWS_PER_THREAD - 1) / ROWS_PER_THREAD;
    int blocks = (int)((totalThreads + BLOCK - 1) / BLOCK); // 1024 for N=2M
    long long stride = (long long)blocks * BLOCK;

    float* partial = (float*)d_ws;
    int*   pflag   = (int*)((char*)d_ws + (size_t)blocks * sizeof(float));

    ce_partial_kernel<<<blocks, BLOCK, 0, stream>>>(logits, target, partial, pflag,
                                                    nrows, stride);
    ce_final_kernel<<<1, BLOCK, 0, stream>>>(partial, pflag, blocks, epoch,
                                             (float*)d_out, nrows);
}